// EfficientInteractionHadamard_78408922956491
// MI455X (gfx1250) — compile-verified
//
#include <hip/hip_runtime.h>

typedef __attribute__((ext_vector_type(2))) float v2f;
typedef __attribute__((ext_vector_type(8))) float v8f;

// D(16x16,f32) = A(16x4,f32) * B(4x16,f32) + C ; wave32, 8-arg VOP3P form
__device__ __forceinline__ v8f wmma_f32(v2f a, v2f b, v8f c) {
    return __builtin_amdgcn_wmma_f32_16x16x4_f32(
        /*neg_a=*/false, a, /*neg_b=*/false, b,
        /*c_mod=*/(short)0, c, /*reuse_a=*/false, /*reuse_b=*/false);
}

// Kernel 1: recover per-edge [start, count) from sorted (id_reduce, Kidx).
// Each edge has exactly one triplet with Kidx==0 (its start) and exactly one
// run-end triplet (its count-1) -> no write races, fully deterministic.
__global__ void build_edge_index(const int* __restrict__ id_reduce,
                                 const int* __restrict__ Kidx,
                                 int nT,
                                 int* __restrict__ edge_start,
                                 int* __restrict__ edge_count) {
    int t = blockIdx.x * blockDim.x + threadIdx.x;
    if (t >= nT) return;
    int e = id_reduce[t];
    int k = Kidx[t];
    if (k == 0) edge_start[e] = t;
    if (t == nT - 1 || id_reduce[t + 1] != e) edge_count[e] = k + 1;
}

// Kernel 2: one wave32 per edge.
//  sum_k (16x64)  = sph_e(16x24) @ M_e(24x64)        [24 wmma]
//  WRt   (16x64)  = rbf_e^T(16x64) @ W^T(64x64)      [64 wmma]
//  m_ca[d]        = sum_s WRt[s,d] * sum_k[s,d]      [8 FMA + shfl_xor(16)]
// All B loads are unconditional (row index clamped into the valid range);
// the scatter's implicit zero rows are realized by zeroing the A (sph)
// element for k >= count via branch-free v_cndmask, keeping EXEC intact and
// letting all loads of a k-step issue together.
__launch_bounds__(128)
__global__ void edge_interaction_kernel(const float* __restrict__ rbf_W1, // (E,64,16)
                                        const float* __restrict__ sph,    // (E,16,24)
                                        const float* __restrict__ m,      // (nT,64)
                                        const float* __restrict__ weight, // (64,64) = weight[:,0,:]
                                        const int*   __restrict__ edge_start,
                                        const int*   __restrict__ edge_count,
                                        float* __restrict__ out,          // (E,64)
                                        int nEdges) {
    const int wave = threadIdx.x >> 5;
    const int e    = blockIdx.x * 4 + wave;
    if (e >= nEdges) return;                 // wave-uniform: EXEC stays all-ones

    const int lane  = threadIdx.x & 31;
    const int row16 = lane & 15;             // M index (0..15) within tile
    const int koff  = (lane >> 4) << 1;      // K offset: lanes 0-15 -> 0, 16-31 -> 2

    const float* __restrict__ sphE = sph    + (size_t)e * (16 * 24);
    const float* __restrict__ rbfE = rbf_W1 + (size_t)e * (64 * 16);
    const int start = edge_start[e];
    const int count = edge_count[e];         // >= 1 always

    // ---- G1: sum_k[s,d] = sum_k sph[e,s,k] * M[k,d], M rows from m[start+k] ----
    v8f sk[4] = {v8f(0.f), v8f(0.f), v8f(0.f), v8f(0.f)};
    #pragma unroll
    for (int kk = 0; kk < 6; ++kk) {
        const int k0 = 4 * kk + koff;        // even; k0,k0+1 contiguous
        const int k1 = k0 + 1;

        // A pair: one 8B-aligned b64 load, then branch-free zeroing
        const v2f araw = *(const v2f*)(sphE + row16 * 24 + k0);
        v2f a;
        a.x = (k0 < count) ? araw.x : 0.0f;  // v_cndmask, no exec change
        a.y = (k1 < count) ? araw.y : 0.0f;

        // B rows: clamped, unconditional, streamed (bypass-ish caching)
        const int kc0 = (k0 < count) ? k0 : (count - 1);
        const int kc1 = (k1 < count) ? k1 : (count - 1);
        const float* __restrict__ m0 = m + (size_t)(start + kc0) * 64;
        const float* __restrict__ m1 = m + (size_t)(start + kc1) * 64;

        v2f b[4];
        #pragma unroll
        for (int nt = 0; nt < 4; ++nt) {
            const int d = nt * 16 + row16;
            b[nt].x = __builtin_nontemporal_load(m0 + d);
            b[nt].y = __builtin_nontemporal_load(m1 + d);
        }
        #pragma unroll
        for (int nt = 0; nt < 4; ++nt) {
            sk[nt] = wmma_f32(a, b[nt], sk[nt]);
        }
    }

    // ---- G2: WRt[s,d] = sum_i rbf[e,i,s] * W[d,i]  (same D layout as sk) ----
    v8f wr[4] = {v8f(0.f), v8f(0.f), v8f(0.f), v8f(0.f)};
    #pragma unroll
    for (int kk = 0; kk < 16; ++kk) {
        const int i0 = 4 * kk + koff;        // even
        v2f a;
        a.x = rbfE[i0 * 16 + row16];         // rbf[e, i0,   s=row16]
        a.y = rbfE[(i0 + 1) * 16 + row16];   // rbf[e, i0+1, s=row16]

        v2f b[4];
        #pragma unroll
        for (int nt = 0; nt < 4; ++nt) {
            const int d = nt * 16 + row16;
            b[nt] = *(const v2f*)(weight + d * 64 + i0);  // W hot in L0/L2
        }
        #pragma unroll
        for (int nt = 0; nt < 4; ++nt) {
            wr[nt] = wmma_f32(a, b[nt], wr[nt]);
        }
    }

    // ---- Final: m_ca[d] = sum_{s=0..15} WRt[s,d]*sum_k[s,d] ----
    // Lane L holds s = v + (L>=16 ? 8 : 0), d = nt*16 + (L&15); lanes L and
    // L+16 share d -> one shfl_xor(16) completes the 16-term reduction.
    #pragma unroll
    for (int nt = 0; nt < 4; ++nt) {
        float p = 0.0f;
        #pragma unroll
        for (int v = 0; v < 8; ++v) p += wr[nt][v] * sk[nt][v];
        p += __shfl_xor(p, 16, 32);
        if (lane < 16)
            __builtin_nontemporal_store(p, out + (size_t)e * 64 + nt * 16 + lane);
    }
}

extern "C" void kernel_launch(void* const* d_in, const int* in_sizes, int n_in,
                              void* d_out, int out_size, void* d_ws, size_t ws_size,
                              hipStream_t stream) {
    const float* rbf_W1   = (const float*)d_in[0];  // (E,64,16)
    const float* sph      = (const float*)d_in[1];  // (E,16,24)
    const float* m        = (const float*)d_in[2];  // (nT,64)
    const float* weight   = (const float*)d_in[3];  // (64,1,64)
    const int*   id_reduce= (const int*)  d_in[4];  // (nT,)
    const int*   Kidx     = (const int*)  d_in[5];  // (nT,)
    float*       out      = (float*)d_out;          // (E,64)

    const int nT     = in_sizes[4];
    const int nEdges = out_size / 64;

    int* edge_start = (int*)d_ws;
    int* edge_count = edge_start + nEdges;

    build_edge_index<<<(nT + 255) / 256, 256, 0, stream>>>(
        id_reduce, Kidx, nT, edge_start, edge_count);

    edge_interaction_kernel<<<(nEdges + 3) / 4, 128, 0, stream>>>(
        rbf_W1, sph, m, weight, edge_start, edge_count, out, nEdges);
}